// DLPerGene_5480378270212
// MI455X (gfx1250) — compile-verified
//
#include <hip/hip_runtime.h>

#define BATCH   1024
#define IN_DIM  20128
#define OUT_DIM 20000
#define KDIM    128     // IN_DIM - OUT_DIM

#define WAVES_PER_BLOCK 8
#define N_PER_WAVE      64   // 4 WMMA tiles of 16 columns
#define N_PER_BLOCK     (WAVES_PER_BLOCK * N_PER_WAVE)  // 512
#define M_PER_BLOCK     16

#define KSTEPS_PER_CHUNK 4   // 4 k-steps of K=4 -> K=16 per chunk
#define NCHUNKS (KDIM / (4 * KSTEPS_PER_CHUNK))          // 8

typedef float v2f __attribute__((ext_vector_type(2)));
typedef float v8f __attribute__((ext_vector_type(8)));
// dword-aligned vector view (hardware DWORD alignment mode handles b64 @4B)
typedef v2f u2f __attribute__((aligned(4)));

__global__ __launch_bounds__(256) void dlpergene_wmma_kernel(
    const float* __restrict__ x,   // (BATCH, IN_DIM)
    const float* __restrict__ W,   // (OUT_DIM, 1 + KDIM) row-major
    const float* __restrict__ b,   // (OUT_DIM)
    float* __restrict__ out)       // (BATCH, OUT_DIM)
{
  const int lane = threadIdx.x & 31;
  const int wave = threadIdx.x >> 5;
  const int l16  = lane & 15;
  const int half = lane >> 4;          // 0: lanes 0-15, 1: lanes 16-31

  const int m0 = blockIdx.y * M_PER_BLOCK;
  const int n0 = blockIdx.x * N_PER_BLOCK + wave * N_PER_WAVE;

  // A fragment source: row = m0 + l16, nongene columns x[row][OUT_DIM + k]
  const float* arow = x + (size_t)(m0 + l16) * IN_DIM + OUT_DIM;

  // B fragment source: column n (clamped for edge tiles so EXEC stays all-1s),
  // shared_w[n][k] = W[n*129 + 1 + k]
  int ncol[4];
  const float* wp[4];
#pragma unroll
  for (int t = 0; t < 4; ++t) {
    int n = n0 + t * 16 + l16;
    ncol[t] = (n < OUT_DIM) ? n : (OUT_DIM - 1);
    wp[t] = W + (size_t)ncol[t] * (1 + KDIM) + 1;
  }

  v8f acc[4];
#pragma unroll
  for (int t = 0; t < 4; ++t) acc[t] = (v8f){0.f,0.f,0.f,0.f,0.f,0.f,0.f,0.f};

  // ---- Software-pipelined K loop: double-buffered fragment chunks ----
  // Per-lane K slice of a k-step at K base k: elements k+2*half, k+2*half+1.
  v2f abuf[2][KSTEPS_PER_CHUNK];
  v2f bbuf[2][KSTEPS_PER_CHUNK][4];

  auto load_chunk = [&](int c, int buf) {
#pragma unroll
    for (int s = 0; s < KSTEPS_PER_CHUNK; ++s) {
      const int kk = (c * KSTEPS_PER_CHUNK + s) * 4 + 2 * half;
      abuf[buf][s] = *(const u2f*)(arow + kk);
#pragma unroll
      for (int t = 0; t < 4; ++t)
        bbuf[buf][s][t] = *(const u2f*)(wp[t] + kk);
    }
  };

  load_chunk(0, 0);
#pragma unroll
  for (int c = 0; c < NCHUNKS; ++c) {
    const int cur = c & 1;
    const int nxt = cur ^ 1;
    if (c < NCHUNKS - 1) load_chunk(c + 1, nxt);   // prefetch burst (20 loads)
#pragma unroll
    for (int s = 0; s < KSTEPS_PER_CHUNK; ++s) {
#pragma unroll
      for (int t = 0; t < 4; ++t) {
        acc[t] = __builtin_amdgcn_wmma_f32_16x16x4_f32(
            /*neg_a=*/false, abuf[cur][s],
            /*neg_b=*/false, bbuf[cur][s][t],
            /*c_mod=*/(short)0, acc[t],
            /*reuse_a=*/false, /*reuse_b=*/false);
      }
    }
  }

  // ---- Fused epilogue: y = acc + gene*diag_w + b ----
  // C layout: VGPR r -> M = r + 8*half, N = l16 (per tile).
  float dw[4], bv[4];
#pragma unroll
  for (int t = 0; t < 4; ++t) {
    dw[t] = W[(size_t)ncol[t] * (1 + KDIM)];  // diag weight
    bv[t] = b[ncol[t]];
  }

  // Burst all gene loads (each element touched exactly once -> non-temporal)
  float gbuf[4][8];
#pragma unroll
  for (int t = 0; t < 4; ++t) {
#pragma unroll
    for (int r = 0; r < 8; ++r) {
      const int row = m0 + r + 8 * half;
      gbuf[t][r] = __builtin_nontemporal_load(x + (size_t)row * IN_DIM + ncol[t]);
    }
  }

#pragma unroll
  for (int t = 0; t < 4; ++t) {
    const int n = n0 + t * 16 + l16;
    const bool valid = (n < OUT_DIM);
#pragma unroll
    for (int r = 0; r < 8; ++r) {
      const int row = m0 + r + 8 * half;
      const float v = acc[t][r] + gbuf[t][r] * dw[t] + bv[t];
      if (valid) {
        __builtin_nontemporal_store(v, out + (size_t)row * OUT_DIM + n);
      }
    }
  }
}

extern "C" void kernel_launch(void* const* d_in, const int* in_sizes, int n_in,
                              void* d_out, int out_size, void* d_ws, size_t ws_size,
                              hipStream_t stream) {
  const float* x = (const float*)d_in[0];
  const float* W = (const float*)d_in[1];
  const float* b = (const float*)d_in[2];
  float* out = (float*)d_out;

  dim3 grid((OUT_DIM + N_PER_BLOCK - 1) / N_PER_BLOCK,  // 40
            BATCH / M_PER_BLOCK);                        // 64
  dlpergene_wmma_kernel<<<grid, WAVES_PER_BLOCK * 32, 0, stream>>>(x, W, b, out);
}